// GATEncoder_38886633898506
// MI455X (gfx1250) — compile-verified
//
#include <hip/hip_runtime.h>
#include <hip/hip_bf16.h>

// ---------------------------------------------------------------------------
// GAT encoder, 3 layers, HID=64, HEADS=4, C=16, for MI455X (gfx1250, wave32)
// ---------------------------------------------------------------------------

typedef float v2f __attribute__((ext_vector_type(2)));
typedef float v8f __attribute__((ext_vector_type(8)));

#define HID 64
#define HEADS 4
#define CPH 16
#define NEG_SLOPE 0.2f
#define LN_EPS 1e-5f

static __host__ __device__ inline int cdiv_i(int a, int b) { return (a + b - 1) / b; }

// Non-returning, relaxed, agent-scope float atomic add -> native
// GLOBAL_ATOMIC_ADD_F32 on CDNA5 (no CAS loop).
__device__ inline void atomicAddF(float* p, float v) {
  __hip_atomic_fetch_add(p, v, __ATOMIC_RELAXED, __HIP_MEMORY_SCOPE_AGENT);
}

// order-preserving float <-> uint mapping for atomic max
__device__ inline unsigned fmap(float f) {
  unsigned u = __float_as_uint(f);
  return (u & 0x80000000u) ? ~u : (u | 0x80000000u);
}
__device__ inline float funmap(unsigned m) {
  unsigned u = (m & 0x80000000u) ? (m ^ 0x80000000u) : ~m;
  return __uint_as_float(u);
}

// ---------------------------------------------------------------------------
// zero fill
__global__ __launch_bounds__(256) void zero_k(float* __restrict__ p, long long count) {
  long long i = (long long)blockIdx.x * 256 + threadIdx.x;
  if (i < count) p[i] = 0.0f;
}

// ---------------------------------------------------------------------------
// tiny precompute: W_ea[3][64][4], M[3][4][4], eb_att[3][4]
__global__ __launch_bounds__(768) void precompute_small(
    const float* __restrict__ lin_edge_w, // [3][64][64]
    const float* __restrict__ att_edge,   // [3][4][16]
    const float* __restrict__ edge_w,     // [4][64]
    const float* __restrict__ edge_b,     // [64]
    float* __restrict__ W_ea,             // [3][64][4]
    float* __restrict__ Mm,               // [3][4][4]
    float* __restrict__ eb_att)           // [3][4]
{
  int t = threadIdx.x;
  if (t < 768) {
    int i = t >> 8, rem = t & 255, k = rem >> 2, h = rem & 3;
    float s = 0.f;
    #pragma unroll
    for (int c = 0; c < CPH; ++c)
      s += lin_edge_w[(i * 64 + k) * 64 + h * CPH + c] * att_edge[(i * 4 + h) * CPH + c];
    W_ea[i * 256 + k * 4 + h] = s;
  }
  __syncthreads();
  if (t < 48) {
    int i = t / 16, rem = t % 16, j = rem >> 2, h = rem & 3;
    float s = 0.f;
    for (int k = 0; k < 64; ++k) s += edge_w[j * 64 + k] * W_ea[i * 256 + k * 4 + h];
    Mm[(i * 4 + j) * 4 + h] = s;
  } else if (t < 60) {
    int u = t - 48, i = u >> 2, h = u & 3;
    float s = 0.f;
    for (int k = 0; k < 64; ++k) s += edge_b[k] * W_ea[i * 256 + k * 4 + h];
    eb_att[i * 4 + h] = s;
  }
}

// ---------------------------------------------------------------------------
// h0 = x @ node_w + node_b   ([N,8]x[8,64], memory bound)
__global__ __launch_bounds__(256) void node_proj(
    const float* __restrict__ x, const float* __restrict__ w,
    const float* __restrict__ b, float* __restrict__ h, int n)
{
  __shared__ float sw[8 * 64];
  __shared__ float sb[64];
  for (int i = threadIdx.x; i < 512; i += 256) sw[i] = w[i];
  if (threadIdx.x < 64) sb[threadIdx.x] = b[threadIdx.x];
  __syncthreads();
  int idx = blockIdx.x * 256 + threadIdx.x;
  if (idx >= n * 64) return;
  int node = idx >> 6, ch = idx & 63;
  const float* xr = x + node * 8;
  float s = sb[ch];
  #pragma unroll
  for (int j = 0; j < 8; ++j) s += xr[j] * sw[j * 64 + ch];
  h[idx] = s;
}

// ---------------------------------------------------------------------------
// deg[n] += 1, attr_sum[n][0..3] += edge_attr[e]  (per real edge, over dst)
__global__ __launch_bounds__(256) void deg_attr_k(
    const int* __restrict__ dst, const float* __restrict__ eattr,
    float* __restrict__ deg, float* __restrict__ asum, int E_)
{
  int e = blockIdx.x * 256 + threadIdx.x;
  if (e >= E_) return;
  int d = dst[e];
  atomicAddF(&deg[d], 1.0f);
  #pragma unroll
  for (int j = 0; j < 4; ++j) atomicAddF(&asum[d * 4 + j], eattr[e * 4 + j]);
}

// ---------------------------------------------------------------------------
// a_loop[i][n][h] = e_mean[n] @ W_ea[i],  e_mean from attr_sum/deg (all linear)
__global__ __launch_bounds__(128) void aloop_k(
    const float* __restrict__ asum, const float* __restrict__ deg,
    const float* __restrict__ edge_w, const float* __restrict__ edge_b,
    const float* __restrict__ W_ea, float* __restrict__ a_loop, int n)
{
  __shared__ float sew[4 * 64];
  __shared__ float seb[64];
  __shared__ float swea[3 * 64 * 4];
  for (int i = threadIdx.x; i < 256; i += 128) sew[i] = edge_w[i];
  for (int i = threadIdx.x; i < 64; i += 128) seb[i] = edge_b[i];
  for (int i = threadIdx.x; i < 768; i += 128) swea[i] = W_ea[i];
  __syncthreads();
  int node = blockIdx.x * 128 + threadIdx.x;
  if (node >= n) return;
  float a0 = asum[node * 4 + 0], a1 = asum[node * 4 + 1];
  float a2 = asum[node * 4 + 2], a3 = asum[node * 4 + 3];
  float dg = deg[node];
  float inv = 1.0f / fmaxf(dg, 1.0f);
  float acc[12];
  #pragma unroll
  for (int u = 0; u < 12; ++u) acc[u] = 0.f;
  for (int k = 0; k < 64; ++k) {
    float em = (a0 * sew[k] + a1 * sew[64 + k] + a2 * sew[128 + k] +
                a3 * sew[192 + k] + dg * seb[k]) * inv;
    #pragma unroll
    for (int u = 0; u < 12; ++u)
      acc[u] += em * swea[(u >> 2) * 256 + k * 4 + (u & 3)];
  }
  #pragma unroll
  for (int u = 0; u < 12; ++u)
    a_loop[(size_t)(u >> 2) * n * 4 + (size_t)node * 4 + (u & 3)] = acc[u];
}

// ---------------------------------------------------------------------------
// xs = h @ W  ([N,64]x[64,64]) via V_WMMA_F32_16X16X4_F32.
// One wave: 16-row tile x all 64 cols; 16 K-steps x 4 col-tiles = 64 WMMAs.
__global__ __launch_bounds__(256) void gemm_n64(
    const float* __restrict__ H, const float* __restrict__ W,
    float* __restrict__ XS, int nrows)
{
  // W transposed into LDS: Wt[col][k]; row stride 66 keeps 8B alignment (k even)
  __shared__ float Wt[64 * 66];
  for (int idx = threadIdx.x; idx < 64 * 64; idx += 256) {
    int k = idx >> 6, c = idx & 63;
    Wt[c * 66 + k] = W[idx];
  }
  __syncthreads();
  int wave = threadIdx.x >> 5;
  int lane = threadIdx.x & 31;
  int tile = blockIdx.x * 8 + wave;
  int row0 = tile * 16;
  if (row0 >= nrows) return;
  int m = lane & 15, half = lane >> 4;
  const float* hrow = H + (size_t)(row0 + m) * 64;
  v8f c0 = {}, c1 = {}, c2 = {}, c3 = {};
  #pragma unroll
  for (int ks = 0; ks < 16; ++ks) {
    int k = ks * 4 + half * 2;                    // ISA 16x4 A layout: VGPR0=K0/K2
    v2f a  = *(const v2f*)(hrow + k);
    v2f b0 = *(const v2f*)(&Wt[(m +  0) * 66 + k]);
    v2f b1 = *(const v2f*)(&Wt[(m + 16) * 66 + k]);
    v2f b2 = *(const v2f*)(&Wt[(m + 32) * 66 + k]);
    v2f b3 = *(const v2f*)(&Wt[(m + 48) * 66 + k]);
    c0 = __builtin_amdgcn_wmma_f32_16x16x4_f32(false, a, false, b0, (short)0, c0, false, false);
    c1 = __builtin_amdgcn_wmma_f32_16x16x4_f32(false, a, false, b1, (short)0, c1, false, false);
    c2 = __builtin_amdgcn_wmma_f32_16x16x4_f32(false, a, false, b2, (short)0, c2, false, false);
    c3 = __builtin_amdgcn_wmma_f32_16x16x4_f32(false, a, false, b3, (short)0, c3, false, false);
  }
  float* out = XS + (size_t)row0 * 64;
  #pragma unroll
  for (int g = 0; g < 8; ++g) {
    int r = g + 8 * half;                         // C layout: VGPR g -> rows g / g+8
    out[(size_t)r * 64 +  0 + m] = c0[g];
    out[(size_t)r * 64 + 16 + m] = c1[g];
    out[(size_t)r * 64 + 32 + m] = c2[g];
    out[(size_t)r * 64 + 48 + m] = c3[g];
  }
}

// ---------------------------------------------------------------------------
// a_src/a_dst[n][h] from xs and per-layer attention vectors
__global__ __launch_bounds__(256) void att_node(
    const float* __restrict__ xs, const float* __restrict__ att_s,
    const float* __restrict__ att_d, float* __restrict__ a_src,
    float* __restrict__ a_dst, int n)
{
  int idx = blockIdx.x * 256 + threadIdx.x;
  if (idx >= n * 4) return;
  int node = idx >> 2, h = idx & 3;
  const float* xr = xs + (size_t)node * 64 + h * CPH;
  float s1 = 0.f, s2 = 0.f;
  #pragma unroll
  for (int c = 0; c < CPH; ++c) {
    float v = xr[c];
    s1 += v * att_s[h * CPH + c];
    s2 += v * att_d[h * CPH + c];
  }
  a_src[idx] = s1;
  a_dst[idx] = s2;
}

// ---------------------------------------------------------------------------
// pass 1: alpha = leaky_relu(a_src[s]+a_dst[d]+a_edge); segment max via atomicMax
__global__ __launch_bounds__(256) void edge_alpha(
    const int* __restrict__ src, const int* __restrict__ dst,
    const float* __restrict__ eattr, const float* __restrict__ Mm,
    const float* __restrict__ ebatt, const float* __restrict__ a_loop_l,
    const float* __restrict__ a_src, const float* __restrict__ a_dst,
    float* __restrict__ alpha, unsigned* __restrict__ amax, int E_, int EP)
{
  int e = blockIdx.x * 256 + threadIdx.x;
  if (e >= EP) return;
  int s, d;
  float ae[4];
  if (e < E_) {
    s = src[e];
    d = dst[e];
    float t0 = eattr[e * 4 + 0], t1 = eattr[e * 4 + 1];
    float t2 = eattr[e * 4 + 2], t3 = eattr[e * 4 + 3];
    #pragma unroll
    for (int h = 0; h < 4; ++h)
      ae[h] = ebatt[h] + t0 * Mm[h] + t1 * Mm[4 + h] + t2 * Mm[8 + h] + t3 * Mm[12 + h];
  } else {
    s = d = e - E_;
    #pragma unroll
    for (int h = 0; h < 4; ++h) ae[h] = a_loop_l[(size_t)s * 4 + h];
  }
  #pragma unroll
  for (int h = 0; h < 4; ++h) {
    float al = a_src[(size_t)s * 4 + h] + a_dst[(size_t)d * 4 + h] + ae[h];
    al = (al > 0.f) ? al : NEG_SLOPE * al;
    alpha[(size_t)e * 4 + h] = al;
    atomicMax(&amax[(size_t)d * 4 + h], fmap(al));
  }
}

// ---------------------------------------------------------------------------
// pass 2: ex = exp(alpha - amax[dst]); denom[dst] += ex; alpha overwritten w/ ex
__global__ __launch_bounds__(256) void edge_sum(
    const int* __restrict__ dst, float* __restrict__ alpha,
    const unsigned* __restrict__ amax, float* __restrict__ denom, int E_, int EP)
{
  int e = blockIdx.x * 256 + threadIdx.x;
  if (e >= EP) return;
  int d = (e < E_) ? dst[e] : (e - E_);
  #pragma unroll
  for (int h = 0; h < 4; ++h) {
    float mx = funmap(amax[(size_t)d * 4 + h]);
    float ex = __expf(alpha[(size_t)e * 4 + h] - mx);
    alpha[(size_t)e * 4 + h] = ex;
    atomicAddF(&denom[(size_t)d * 4 + h], ex);
  }
}

// ---------------------------------------------------------------------------
// pass 3: accum[dst][ch] += xs[src][ch] * ex/denom  (thread per (edge, channel))
__global__ __launch_bounds__(256) void edge_scatter(
    const int* __restrict__ src, const int* __restrict__ dst,
    const float* __restrict__ ex, const float* __restrict__ denom,
    const float* __restrict__ xs, float* __restrict__ accum, int E_, int EP)
{
  long long idx = (long long)blockIdx.x * 256 + threadIdx.x;
  if (idx >= (long long)EP * 64) return;
  int e = (int)(idx >> 6);
  int ch = (int)(idx & 63);
  int h = ch >> 4;
  int s = (e < E_) ? src[e] : (e - E_);
  int d = (e < E_) ? dst[e] : (e - E_);
  float w = ex[(size_t)e * 4 + h] / (denom[(size_t)d * 4 + h] + 1e-16f);
  atomicAddF(&accum[(size_t)d * 64 + ch], xs[(size_t)s * 64 + ch] * w);
}

// ---------------------------------------------------------------------------
// h_next = relu(layer_norm(accum + gat_bias + residual))  (wave32 per node)
__global__ __launch_bounds__(256) void norm_k(
    const float* __restrict__ accum, const float* __restrict__ res,
    const float* __restrict__ bias, const float* __restrict__ gamma,
    const float* __restrict__ beta, float* __restrict__ out, int n)
{
  int wave = threadIdx.x >> 5, lane = threadIdx.x & 31;
  int node = blockIdx.x * 8 + wave;
  if (node >= n) return;
  size_t base = (size_t)node * 64;
  float v0 = accum[base + lane]      + bias[lane]      + res[base + lane];
  float v1 = accum[base + 32 + lane] + bias[32 + lane] + res[base + 32 + lane];
  float s = v0 + v1;
  #pragma unroll
  for (int m = 16; m > 0; m >>= 1) s += __shfl_xor(s, m, 32);
  float mu = s * (1.0f / 64.0f);
  float d0 = v0 - mu, d1 = v1 - mu;
  float q = d0 * d0 + d1 * d1;
  #pragma unroll
  for (int m = 16; m > 0; m >>= 1) q += __shfl_xor(q, m, 32);
  float r = rsqrtf(q * (1.0f / 64.0f) + LN_EPS);
  float o0 = d0 * r * gamma[lane]      + beta[lane];
  float o1 = d1 * r * gamma[32 + lane] + beta[32 + lane];
  out[base + lane]      = fmaxf(o0, 0.0f);
  out[base + 32 + lane] = fmaxf(o1, 0.0f);
}

// ---------------------------------------------------------------------------
extern "C" void kernel_launch(void* const* d_in, const int* in_sizes, int n_in,
                              void* d_out, int out_size, void* d_ws, size_t ws_size,
                              hipStream_t stream) {
  const float* x          = (const float*)d_in[0];
  const int*   edge_index = (const int*)  d_in[1];
  const float* edge_attr  = (const float*)d_in[2];
  const float* node_w     = (const float*)d_in[3];
  const float* node_b     = (const float*)d_in[4];
  const float* edge_w     = (const float*)d_in[5];
  const float* edge_b     = (const float*)d_in[6];
  const float* lin_w      = (const float*)d_in[7];
  const float* lin_edge_w = (const float*)d_in[8];
  const float* att_src    = (const float*)d_in[9];
  const float* att_dst    = (const float*)d_in[10];
  const float* att_edge   = (const float*)d_in[11];
  const float* gat_bias   = (const float*)d_in[12];
  const float* ln_gamma   = (const float*)d_in[13];
  const float* ln_beta    = (const float*)d_in[14];
  float* outp = (float*)d_out;

  const int N_ = in_sizes[0] / 8;     // 100000
  const int E_ = in_sizes[1] / 2;     // 1600000
  const int EP = E_ + N_;             // edges + self loops
  const int* srcp = edge_index;
  const int* dstp = edge_index + E_;

  // workspace layout (floats)
  float* ws = (float*)d_ws;
  size_t o = 0;
  float*    hA     = ws + o; o += (size_t)N_ * 64;
  float*    xs     = ws + o; o += (size_t)N_ * 64;
  float*    alpha  = ws + o; o += (size_t)EP * 4;
  float*    a_src  = ws + o; o += (size_t)N_ * 4;
  float*    a_dst  = ws + o; o += (size_t)N_ * 4;
  float*    a_loop = ws + o; o += (size_t)3 * N_ * 4;
  float*    amaxf  = ws + o; o += (size_t)N_ * 4;   // contiguous zero region start
  float*    denom  = ws + o; o += (size_t)N_ * 4;
  float*    accum  = ws + o; o += (size_t)N_ * 64;
  float*    deg    = ws + o; o += (size_t)N_;       // contiguous zero region start
  float*    asum   = ws + o; o += (size_t)N_ * 4;
  float*    W_ea   = ws + o; o += 3 * 64 * 4;
  float*    Mm     = ws + o; o += 3 * 4 * 4;
  float*    ebatt  = ws + o; o += 3 * 4;
  unsigned* amax   = (unsigned*)amaxf;

  // ---- setup (once per call) ----
  {
    long long cnt = (long long)N_ * 5;            // deg + asum
    zero_k<<<(int)((cnt + 255) / 256), 256, 0, stream>>>(deg, cnt);
  }
  precompute_small<<<1, 768, 0, stream>>>(lin_edge_w, att_edge, edge_w, edge_b,
                                          W_ea, Mm, ebatt);
  node_proj<<<cdiv_i(N_ * 64, 256), 256, 0, stream>>>(x, node_w, node_b, hA, N_);
  deg_attr_k<<<cdiv_i(E_, 256), 256, 0, stream>>>(dstp, edge_attr, deg, asum, E_);
  aloop_k<<<cdiv_i(N_, 128), 128, 0, stream>>>(asum, deg, edge_w, edge_b, W_ea,
                                               a_loop, N_);

  // ---- layers ----
  for (int i = 0; i < 3; ++i) {
    float* cur = (i % 2 == 0) ? hA : outp;   // current h (residual source)
    float* nxt = (i % 2 == 0) ? outp : hA;   // next h (layer 2 -> d_out)

    long long zcnt = (long long)N_ * 72;     // amax + denom + accum, contiguous
    zero_k<<<(int)((zcnt + 255) / 256), 256, 0, stream>>>(amaxf, zcnt);

    gemm_n64<<<cdiv_i(cdiv_i(N_, 16), 8), 256, 0, stream>>>(
        cur, lin_w + (size_t)i * 64 * 64, xs, N_);

    att_node<<<cdiv_i(N_ * 4, 256), 256, 0, stream>>>(
        xs, att_src + i * 64, att_dst + i * 64, a_src, a_dst, N_);

    edge_alpha<<<cdiv_i(EP, 256), 256, 0, stream>>>(
        srcp, dstp, edge_attr, Mm + i * 16, ebatt + i * 4,
        a_loop + (size_t)i * N_ * 4, a_src, a_dst, alpha, amax, E_, EP);

    edge_sum<<<cdiv_i(EP, 256), 256, 0, stream>>>(dstp, alpha, amax, denom, E_, EP);

    long long scnt = (long long)EP * 64;
    edge_scatter<<<(int)((scnt + 255) / 256), 256, 0, stream>>>(
        srcp, dstp, alpha, denom, xs, accum, E_, EP);

    norm_k<<<cdiv_i(N_, 8), 256, 0, stream>>>(
        accum, cur, gat_bias + i * 64, ln_gamma + i * 64, ln_beta + i * 64, nxt, N_);
  }
}